// CausalUserEncoder_43361989820607
// MI455X (gfx1250) — compile-verified
//
#include <hip/hip_runtime.h>
#include <hip/hip_bf16.h>
#include <math.h>

typedef float v2f __attribute__((ext_vector_type(2)));
typedef float v8f __attribute__((ext_vector_type(8)));

#define BB 16
#define SS 512
#define DD 256
#define HH 16
#define HD 16
#define D3 768
#define SCPAD 516   // 516 % 64 == 4 -> 16 rows hit 16 distinct banks; 516*4 % 16 == 0 -> float4 aligned

__device__ __forceinline__ v8f wmma_f32(v2f a, v2f b, v8f c) {
    // V_WMMA_F32_16X16X4_F32 : D = A(16x4) x B(4x16) + C(16x16), all fp32
    return __builtin_amdgcn_wmma_f32_16x16x4_f32(false, a, false, b, (short)0, c, false, false);
}

// ---------------------------------------------------------------------------
// Kernel 1: x = news + pos[:S]  (elementwise, float4 vectorized)
// ---------------------------------------------------------------------------
__global__ __launch_bounds__(256) void k_addpos(const float* __restrict__ news,
                                                const float* __restrict__ pos,
                                                float* __restrict__ x) {
    int i = (blockIdx.x * 256 + threadIdx.x) * 4;          // 2,097,152 floats exact
    float4 a = *(const float4*)(news + i);
    int p = i & (SS * DD - 1);                             // pos repeats per batch (131072 = 2^17)
    float4 e = *(const float4*)(pos + p);
    float4 r;
    r.x = a.x + e.x; r.y = a.y + e.y; r.z = a.z + e.z; r.w = a.w + e.w;
    *(float4*)(x + i) = r;
}

// ---------------------------------------------------------------------------
// Kernel 2: qkv = x @ in_proj_w^T + in_proj_b   (M=8192, N=768, K=256) fp32 WMMA
// 32x32 tile per wave: 2 A-frags + 2 B-frags feed 4 WMMAs per K-step
// (256 B loaded per wave per WMMA instead of 512 B -> 2x arithmetic intensity).
// ---------------------------------------------------------------------------
__global__ __launch_bounds__(256) void k_qkv(const float* __restrict__ x,
                                             const float* __restrict__ w,
                                             const float* __restrict__ bias,
                                             float* __restrict__ qkv) {
    const int lane = threadIdx.x & 31;
    const int wave = threadIdx.x >> 5;
    const int lm   = lane & 15;
    const int half = lane >> 4;
    const int n0   = blockIdx.x * 32;                      // 24 N tiles of 32
    const int m0   = (blockIdx.y * 8 + wave) * 32;         // 256 M tiles of 32

    const float* xa0 = x + (size_t)(m0 + lm) * DD + 2 * half;       // A rows m0..m0+15
    const float* xa1 = xa0 + (size_t)16 * DD;                       // A rows m0+16..m0+31
    const float* wb0 = w + (size_t)(n0 + lm) * DD + 2 * half;       // B cols n0..n0+15
    const float* wb1 = wb0 + (size_t)16 * DD;                       // B cols n0+16..n0+31

    v8f c00 = {0.f,0.f,0.f,0.f,0.f,0.f,0.f,0.f};
    v8f c01 = c00, c10 = c00, c11 = c00;
#pragma unroll 4
    for (int k0 = 0; k0 < DD; k0 += 4) {
        v2f a0 = *(const v2f*)(xa0 + k0);
        v2f a1 = *(const v2f*)(xa1 + k0);
        v2f b0 = *(const v2f*)(wb0 + k0);
        v2f b1 = *(const v2f*)(wb1 + k0);
        c00 = wmma_f32(a0, b0, c00);
        c01 = wmma_f32(a0, b1, c01);
        c10 = wmma_f32(a1, b0, c10);
        c11 = wmma_f32(a1, b1, c11);
    }
    const float bv0 = bias[n0 + lm];
    const float bv1 = bias[n0 + 16 + lm];
    float* o0 = qkv + (size_t)(m0 + 8 * half) * D3 + n0 + lm;   // rows m0+8h+r
    float* o1 = o0 + (size_t)16 * D3;                            // rows m0+16+8h+r
#pragma unroll
    for (int r = 0; r < 8; ++r) {
        o0[(size_t)r * D3]      = c00[r] + bv0;
        o0[(size_t)r * D3 + 16] = c01[r] + bv1;
        o1[(size_t)r * D3]      = c10[r] + bv0;
        o1[(size_t)r * D3 + 16] = c11[r] + bv1;
    }
}

// ---------------------------------------------------------------------------
// Kernel 3: causal attention per (b, h, 16-row q tile). One wave per block.
// Phase 1: scores (fp32 WMMA) -> masked/scaled into LDS
// Phase 2: softmax (float4 LDS passes, lane-pair combine via shfl_xor 16),
//          coalesced float4 stores of the full [S] weight rows
// Phase 3: attn = P @ V (fp32 WMMA, A-frags straight from LDS), scaled by 1/rowsum
// ---------------------------------------------------------------------------
__global__ __launch_bounds__(32) void k_attn(const float* __restrict__ qkv,
                                             const unsigned char* __restrict__ pad,
                                             float* __restrict__ attnw,   // [B,H,S,S]
                                             float* __restrict__ attn) {  // [B,S,D]
    __shared__ float sc[16][SCPAD];
    __shared__ float rscale[16];

    const int lane = threadIdx.x;
    const int lm   = lane & 15;
    const int half = lane >> 4;
    const int q0   = blockIdx.x * 16;
    const int h    = blockIdx.y;
    const int b    = blockIdx.z;
    const int jmax = q0 >> 4;                 // inclusive diagonal tile

    const float* qbase = qkv + (size_t)(b * SS) * D3 + h * HD;

    // Q A-fragments (reused for every K tile)
    v2f aq[4];
    {
        const float* qp = qbase + (size_t)(q0 + lm) * D3 + 2 * half;
#pragma unroll
        for (int kk = 0; kk < 4; ++kk) aq[kk] = *(const v2f*)(qp + kk * 4);
    }

    // ---- Phase 1: score tiles into LDS ----
    for (int j = 0; j <= jmax; ++j) {
        v8f c = {0.f,0.f,0.f,0.f,0.f,0.f,0.f,0.f};
        const float* kp = qbase + DD + (size_t)(j * 16 + lm) * D3 + 2 * half;
#pragma unroll
        for (int kk = 0; kk < 4; ++kk) {
            v2f bf = *(const v2f*)(kp + kk * 4);
            c = wmma_f32(aq[kk], bf, c);
        }
        const int col = j * 16 + lm;
        const bool padded = pad[b * SS + col] != 0;
#pragma unroll
        for (int r = 0; r < 8; ++r) {
            int row = q0 + r + 8 * half;
            float v = c[r] * 0.25f;                      // 1/sqrt(hd)
            if (padded || col > row) v = -INFINITY;      // causal + padding mask
            sc[r + 8 * half][col] = v;
        }
    }

    const int len = q0 + 16;                 // stored columns; masked ones are -inf

    // ---- Phase 2a: rowmax + exp-sum. Lanes (lm, lm+16) take contiguous row halves
    //      (half-length is a multiple of 8 -> float4-aligned LDS traffic) ----
    {
        const int len2 = len >> 1;
        float* rowp = &sc[lm][half * len2];
        float mx = -INFINITY;
        for (int c = 0; c < len2; c += 4) {
            float4 v4 = *(const float4*)(rowp + c);
            mx = fmaxf(mx, fmaxf(fmaxf(v4.x, v4.y), fmaxf(v4.z, v4.w)));
        }
        mx = fmaxf(mx, __shfl_xor(mx, 16, 32));
        float sum = 0.f;
        for (int c = 0; c < len2; c += 4) {
            float4 v4 = *(float4*)(rowp + c);
            v4.x = __expf(v4.x - mx);
            v4.y = __expf(v4.y - mx);
            v4.z = __expf(v4.z - mx);
            v4.w = __expf(v4.w - mx);
            *(float4*)(rowp + c) = v4;
            sum += (v4.x + v4.y) + (v4.z + v4.w);
        }
        sum += __shfl_xor(sum, 16, 32);
        if (half == 0) rscale[lm] = 1.0f / sum;
    }

    // ---- Phase 2b: write full 512-col weight rows (coalesced float4) ----
    {
        float* wout = attnw + ((size_t)(b * HH + h) * SS + q0) * SS;
        for (int r = 0; r < 16; ++r) {
            const float srow = rscale[r];
            const float* srcrow = sc[r];
            float* orow = wout + (size_t)r * SS;
            for (int c4 = lane * 4; c4 < SS; c4 += 128) {
                float4 wv;
                if (c4 < len) {              // len is a multiple of 16 -> whole f4 in range
                    wv.x = srcrow[c4 + 0] * srow;
                    wv.y = srcrow[c4 + 1] * srow;
                    wv.z = srcrow[c4 + 2] * srow;
                    wv.w = srcrow[c4 + 3] * srow;
                } else {
                    wv.x = wv.y = wv.z = wv.w = 0.f;
                }
                *(float4*)(orow + c4) = wv;
            }
        }
    }

    // ---- Phase 3: attn tile = P(16 x len) @ V(len x 16), fp32 WMMA ----
    {
        v8f acc = {0.f,0.f,0.f,0.f,0.f,0.f,0.f,0.f};
        const float* vbase = qbase + 2 * DD;
        for (int j = 0; j <= jmax; ++j) {
#pragma unroll
            for (int kk = 0; kk < 4; ++kk) {
                const int c0 = j * 16 + kk * 4 + 2 * half;
                v2f a;
                a.x = sc[lm][c0];
                a.y = sc[lm][c0 + 1];
                v2f bv;
                bv.x = vbase[(size_t)c0 * D3 + lm];
                bv.y = vbase[(size_t)(c0 + 1) * D3 + lm];
                acc = wmma_f32(a, bv, acc);
            }
        }
        float* aout = attn + (size_t)(b * SS + q0 + 8 * half) * DD + h * HD + lm;
#pragma unroll
        for (int r = 0; r < 8; ++r) aout[(size_t)r * DD] = acc[r] * rscale[r + 8 * half];
    }
}

// ---------------------------------------------------------------------------
// Kernel 4: only the last valid position per batch matters for `user`.
// out-proj -> proj -> LayerNorm -> ReLU -> L2 normalize. One block per batch.
// ---------------------------------------------------------------------------
__global__ __launch_bounds__(256) void k_final(const float* __restrict__ attn,
                                               const unsigned char* __restrict__ pad,
                                               const float* __restrict__ out_w,
                                               const float* __restrict__ out_b,
                                               const float* __restrict__ proj_w,
                                               const float* __restrict__ proj_b,
                                               const float* __restrict__ ln_g,
                                               const float* __restrict__ ln_b,
                                               float* __restrict__ user) {
    __shared__ float xrow[DD];
    __shared__ float arow[DD];
    __shared__ float red[256];
    __shared__ int s_last;
    const int t = threadIdx.x;
    const int b = blockIdx.x;

    // seq_len = sum(~pad); last = clip(len-1, 0)
    red[t] = (pad[b * SS + t] ? 0.f : 1.f) + (pad[b * SS + 256 + t] ? 0.f : 1.f);
    __syncthreads();
    for (int s = 128; s > 0; s >>= 1) { if (t < s) red[t] += red[t + s]; __syncthreads(); }
    if (t == 0) { int len = (int)red[0]; s_last = len > 0 ? len - 1 : 0; }
    __syncthreads();
    const int last = s_last;

    xrow[t] = attn[(size_t)(b * SS + last) * DD + t];
    __syncthreads();

    // attn_out = xrow @ out_w^T + out_b
    float acc = 0.f;
    {
        const float* wr = out_w + (size_t)t * DD;
        for (int k = 0; k < DD; ++k) acc += xrow[k] * wr[k];
    }
    arow[t] = acc + out_b[t];
    __syncthreads();

    // h = arow @ proj_w^T + proj_b
    float hv = 0.f;
    {
        const float* pr = proj_w + (size_t)t * DD;
        for (int k = 0; k < DD; ++k) hv += arow[k] * pr[k];
    }
    hv += proj_b[t];

    // LayerNorm
    red[t] = hv; __syncthreads();
    for (int s = 128; s > 0; s >>= 1) { if (t < s) red[t] += red[t + s]; __syncthreads(); }
    const float mu = red[0] * (1.0f / DD);
    __syncthreads();
    const float dv = hv - mu;
    red[t] = dv * dv; __syncthreads();
    for (int s = 128; s > 0; s >>= 1) { if (t < s) red[t] += red[t + s]; __syncthreads(); }
    const float var = red[0] * (1.0f / DD);
    __syncthreads();
    float r = dv * rsqrtf(var + 1e-5f) * ln_g[t] + ln_b[t];
    r = fmaxf(r, 0.f);                       // ReLU

    // L2 normalize
    red[t] = r * r; __syncthreads();
    for (int s = 128; s > 0; s >>= 1) { if (t < s) red[t] += red[t + s]; __syncthreads(); }
    const float nrm = sqrtf(red[0]);
    user[b * DD + t] = r / fmaxf(nrm, 1e-12f);
}

// ---------------------------------------------------------------------------
extern "C" void kernel_launch(void* const* d_in, const int* in_sizes, int n_in,
                              void* d_out, int out_size, void* d_ws, size_t ws_size,
                              hipStream_t stream) {
    const float*         news   = (const float*)d_in[0];
    const unsigned char* pad    = (const unsigned char*)d_in[1];  // bool mask
    const float*         pos    = (const float*)d_in[2];
    const float*         in_w   = (const float*)d_in[3];
    const float*         in_b   = (const float*)d_in[4];
    const float*         out_w  = (const float*)d_in[5];
    const float*         out_b  = (const float*)d_in[6];
    const float*         proj_w = (const float*)d_in[7];
    const float*         proj_b = (const float*)d_in[8];
    const float*         ln_g   = (const float*)d_in[9];
    const float*         ln_b   = (const float*)d_in[10];

    float* out   = (float*)d_out;
    float* user  = out;                          // [B, D]
    float* attnw = out + BB * DD;                // [B, H, S, S]

    float* ws   = (float*)d_ws;
    float* x    = ws;                            // 2,097,152 floats
    float* qkv  = ws + 2097152;                  // 6,291,456 floats
    float* attn = ws + 2097152 + 6291456;        // 2,097,152 floats

    k_addpos<<<2048, 256, 0, stream>>>(news, pos, x);
    k_qkv<<<dim3(24, 32), 256, 0, stream>>>(x, in_w, in_b, qkv);
    k_attn<<<dim3(SS / 16, HH, BB), 32, 0, stream>>>(qkv, pad, attnw, attn);
    k_final<<<BB, 256, 0, stream>>>(attn, pad, out_w, out_b, proj_w, proj_b,
                                    ln_g, ln_b, user);
}